// ColorTransformsHSV_1297080123785
// MI455X (gfx1250) — compile-verified
//
#include <hip/hip_runtime.h>

#define RES 72
#define IMG_HW (512 * 512)
#define THREADS 256
#define PIX_PER_THREAD 4
#define PIX_PER_BLOCK (THREADS * PIX_PER_THREAD)   // 1024
#define BLOCKS_PER_IMG (IMG_HW / PIX_PER_BLOCK)    // 256

typedef float v4f __attribute__((ext_vector_type(4)));
typedef int   v4i __attribute__((ext_vector_type(4)));
typedef __attribute__((address_space(1))) v4i* gv4i_ptr;   // global
typedef __attribute__((address_space(3))) v4i* lv4i_ptr;   // LDS

__device__ __forceinline__ float clamp01(float v) {
    return __builtin_fminf(__builtin_fmaxf(v, 0.0f), 1.0f);
}

// LUT lerp: matches reference _lut_transform exactly (floor, frac, clipped idx).
// lut layout in LDS = raw copy of params[n] : [RES][3]
__device__ __forceinline__ float lut_interp(float v, int c, const float* lut) {
    float xs = v * (float)(RES - 1);
    float fl = __builtin_floorf(xs);
    float fp = xs - fl;
    int ip = (int)fl;
    int i0 = ip < 0 ? 0 : (ip > RES - 1 ? RES - 1 : ip);
    int i1 = (ip + 1) < 0 ? 0 : ((ip + 1) > RES - 1 ? RES - 1 : ip + 1);
    float e0 = lut[i0 * 3 + c];
    float e1 = lut[i1 * 3 + c];
    return clamp01(e0 * (1.0f - fp) + e1 * fp);
}

__global__ __launch_bounds__(THREADS) void hsv_lut_rgb_kernel(
    const float* __restrict__ imgs,     // (N,3,512,512)
    const float* __restrict__ params,   // (N,72,3)
    float* __restrict__ out)            // (N,3,512,512)
{
    __shared__ float lut[RES * 3];      // 864 bytes, per-image LUT

    const int n   = blockIdx.x / BLOCKS_PER_IMG;
    const int blk = blockIdx.x % BLOCKS_PER_IMG;
    const float* pImg = params + (size_t)n * (RES * 3);

    // ---- Stage LUT into LDS (CDNA5 async global->LDS) ----
#if __has_builtin(__builtin_amdgcn_global_load_async_to_lds_b128)
    if (threadIdx.x < (RES * 3) / 4) {  // 54 lanes x 16B = 864B
        const float* src = pImg + threadIdx.x * 4;
        __builtin_amdgcn_global_load_async_to_lds_b128(
            (gv4i_ptr)src,
            (lv4i_ptr)&lut[threadIdx.x * 4],
            0, 0);
    }
    asm volatile("s_wait_asynccnt 0" ::: "memory");
#else
    if (threadIdx.x < RES * 3) lut[threadIdx.x] = pImg[threadIdx.x];
#endif
    __syncthreads();

    // ---- Streaming pixel work: one b128 per channel per thread ----
    const size_t base = (size_t)n * 3 * IMG_HW
                      + (size_t)blk * PIX_PER_BLOCK
                      + (size_t)threadIdx.x * PIX_PER_THREAD;

    v4f r = __builtin_nontemporal_load((const v4f*)(imgs + base));
    v4f g = __builtin_nontemporal_load((const v4f*)(imgs + base + IMG_HW));
    v4f b = __builtin_nontemporal_load((const v4f*)(imgs + base + 2 * IMG_HW));

    // constants (double-precision exact, rounded once to f32)
    const float kA  = 0.47140452079103173f;  // sqrt(2)/3
    const float kB  = 0.23570226039551587f;  // 1/(3*sqrt(2))
    const float kC  = 0.40824829046386307f;  // 1/sqrt(6)
    const float kS2 = 1.41421356237309515f;  // sqrt(2)
    const float kIS = 0.70710678118654752f;  // 1/sqrt(2)
    const float kSQ = 1.22474487139158905f;  // sqrt(1.5)

    v4f ro, go, bo;
#pragma unroll
    for (int j = 0; j < PIX_PER_THREAD; ++j) {
        float rv = r[j], gv = g[j], bv = b[j];

        // _from_rgb
        float x = rv * kA - gv * kB - bv * kB;
        float y = (gv - bv) * kC;
        float z = __builtin_fmaxf(rv, __builtin_fmaxf(gv, bv));

        // _lut_transform (per hsv channel)
        float t0 = lut_interp(z,        0, lut);
        float t1 = lut_interp(x + 0.5f, 1, lut);
        float t2 = lut_interp(y + 0.5f, 2, lut);

        // _to_rgb
        float xx = t1 - 0.5f;
        float yy = t2 - 0.5f;
        float rp = kS2 * xx;
        float gp = -xx * kIS + yy * kSQ;
        float bp = -xx * kIS - yy * kSQ;
        float delta = t0 - __builtin_fmaxf(rp, __builtin_fmaxf(gp, bp));

        ro[j] = clamp01(rp + delta);
        go[j] = clamp01(gp + delta);
        bo[j] = clamp01(bp + delta);
    }

    __builtin_nontemporal_store(ro, (v4f*)(out + base));
    __builtin_nontemporal_store(go, (v4f*)(out + base + IMG_HW));
    __builtin_nontemporal_store(bo, (v4f*)(out + base + 2 * IMG_HW));
}

extern "C" void kernel_launch(void* const* d_in, const int* in_sizes, int n_in,
                              void* d_out, int out_size, void* d_ws, size_t ws_size,
                              hipStream_t stream) {
    const float* imgs   = (const float*)d_in[0];
    const float* params = (const float*)d_in[1];
    float* out = (float*)d_out;

    const int N = in_sizes[1] / (RES * 3);   // 32 images
    dim3 grid(N * BLOCKS_PER_IMG);           // 8192 blocks
    dim3 block(THREADS);                     // 256 threads = 8 wave32
    hipLaunchKernelGGL(hsv_lut_rgb_kernel, grid, block, 0, stream,
                       imgs, params, out);
}